// LogEigLayer_18743237279776
// MI455X (gfx1250) — compile-verified
//
#include <hip/hip_runtime.h>
#include <hip/hip_bf16.h>
#include <math.h>

// Batched SPD matrix logarithm: out = log(X + EPS*I) for 8192 matrices of 64x64 fp32.
// Inverse-free, GEMM-only algorithm (Newton-Schulz repeated square roots + Mercator
// series), mapped onto V_WMMA_F32_16X16X4_F32. One block (8 wave32) per matrix.
//
// All intermediate matrices are symmetric (polynomials in the SPD input), so the
// WMMA B-operand fragment B[ka..ka+1][col] == B[col][ka..ka+1] can be read as a
// ROW-contiguous aligned pair: every WMMA operand is a single ds_load_b64, no
// register repacking. Each wave owns two 16x16 tiles in the same row block (shared
// A fragment, two independent accumulator chains).

typedef __attribute__((ext_vector_type(2))) float v2f;
typedef __attribute__((ext_vector_type(8))) float v8f;

#define MATN       64
#define LDS_LD     66           // even pad: 8B-aligned operand pairs, conflict-free banks
#define MAT_ELEMS  (MATN * MATN)
#define EPS_REG    1e-4f
#define NS_ITERS   6            // Newton-Schulz iterations per square root
#define SQRT_STEPS 5            // repeated square roots (2^5 = 32)
#define LOG_SCALE  32.0f

// D = alpha * (A @ B) + diag * I   (all 64x64 symmetric, LDS, leading dim LDS_LD)
// 8 waves: wave w computes row block ti = (w>>1)*16, col blocks tj0 = (w&1)*32, tj0+16.
__device__ __forceinline__ void mm64(const float* __restrict__ A,
                                     const float* __restrict__ B,
                                     float* __restrict__ D,
                                     float alpha, float diag, int lane, int wave) {
  const int m16  = lane & 15;
  const int half = lane >> 4;        // 0: lanes 0-15, 1: lanes 16-31
  const int kb   = half << 1;        // K sub-offset per A/B fragment layout
  const int ti   = (wave >> 1) << 4; // shared row block for both tiles
  const int tj0  = (wave & 1) << 5;  // 0 or 32
  const int tj1  = tj0 + 16;

  const float* arow  = A + (ti  + m16) * LDS_LD + kb;
  const float* brow0 = B + (tj0 + m16) * LDS_LD + kb;  // B symmetric: row == column
  const float* brow1 = B + (tj1 + m16) * LDS_LD + kb;

  v8f acc0 = {0.f, 0.f, 0.f, 0.f, 0.f, 0.f, 0.f, 0.f};
  v8f acc1 = {0.f, 0.f, 0.f, 0.f, 0.f, 0.f, 0.f, 0.f};
#pragma unroll
  for (int k0 = 0; k0 < MATN; k0 += 4) {
    // A 16x4 fragment: lane m16 holds row ti+m16; VGPR0=K(k0+kb), VGPR1=K(k0+kb+1)
    v2f a  = *(const v2f*)(arow  + k0);   // aligned ds_load_b64
    // B 4x16 fragments: rows K(k0+kb), K(k0+kb+1) at column tjX+m16, read via symmetry
    v2f b0 = *(const v2f*)(brow0 + k0);   // aligned ds_load_b64
    v2f b1 = *(const v2f*)(brow1 + k0);   // aligned ds_load_b64
    acc0 = __builtin_amdgcn_wmma_f32_16x16x4_f32(false, a, false, b0, (short)0, acc0,
                                                 false, false);
    acc1 = __builtin_amdgcn_wmma_f32_16x16x4_f32(false, a, false, b1, (short)0, acc1,
                                                 false, false);
  }
  // C/D layout: VGPR r -> rows ti+r (lanes 0-15) and ti+r+8 (lanes 16-31)
  const int rbase = ti + (half << 3);
  const int col0  = tj0 + m16;
  const int col1  = tj1 + m16;
#pragma unroll
  for (int r = 0; r < 8; ++r) {
    const int row = rbase + r;
    float v0 = alpha * acc0[r];
    float v1 = alpha * acc1[r];
    if (row == col0) v0 += diag;
    if (row == col1) v1 += diag;
    D[row * LDS_LD + col0] = v0;
    D[row * LDS_LD + col1] = v1;
  }
}

__global__ __launch_bounds__(256) void logm_spd64_kernel(const float* __restrict__ X,
                                                         float* __restrict__ out,
                                                         int nmat) {
  __shared__ __align__(16) float b0[MATN * LDS_LD];
  __shared__ __align__(16) float b1[MATN * LDS_LD];
  __shared__ __align__(16) float b2[MATN * LDS_LD];
  __shared__ __align__(16) float b3[MATN * LDS_LD];
  __shared__ float red[8];
  __shared__ float scale[2];   // [0] = 1/c, [1] = log(c)

  const int tid  = threadIdx.x;
  const int lane = tid & 31;
  const int wave = tid >> 5;
  const int mat  = blockIdx.x;
  if (mat >= nmat) return;
  const size_t gbase = (size_t)mat * MAT_ELEMS;

  // ---- Load X, add EPS*I, accumulate Frobenius norm^2 ----
  float ssq = 0.f;
  for (int idx = tid; idx < MAT_ELEMS; idx += 256) {
    const int r = idx >> 6, c = idx & 63;
    float v = X[gbase + idx];
    if (r == c) v += EPS_REG;
    b0[r * LDS_LD + c] = v;
    ssq += v * v;
  }
  // wave32 reduction, then cross-wave
#pragma unroll
  for (int off = 16; off > 0; off >>= 1) ssq += __shfl_xor(ssq, off, 32);
  if (lane == 0) red[wave] = ssq;
  __syncthreads();
  if (tid == 0) {
    float s = 0.f;
#pragma unroll
    for (int i = 0; i < 8; ++i) s += red[i];
    const float c = sqrtf(s);
    scale[0] = 1.0f / c;
    scale[1] = logf(c);
  }
  __syncthreads();

  // ---- M = (X + eps I)/c in b0 ; Z = I in b1 ----
  const float inv_c = scale[0];
  for (int idx = tid; idx < MAT_ELEMS; idx += 256) {
    const int r = idx >> 6, c = idx & 63;
    b0[r * LDS_LD + c] *= inv_c;
    b1[r * LDS_LD + c] = (r == c) ? 1.0f : 0.0f;
  }
  __syncthreads();

  float* Y = b0;  // -> M^(1/2^s)
  float* Z = b1;  // -> M^(-1/2^s)
  float* W = b2;
  float* T = b3;

  // ---- Repeated square roots via coupled Newton-Schulz ----
  for (int s = 0; s < SQRT_STEPS; ++s) {
    for (int it = 0; it < NS_ITERS; ++it) {
      mm64(Z, Y, W, -0.5f, 1.5f, lane, wave);   // W = 1.5 I - 0.5 Z Y
      __syncthreads();
      mm64(Y, W, T, 1.0f, 0.0f, lane, wave);    // Ynew = Y W
      __syncthreads();
      mm64(W, Z, Y, 1.0f, 0.0f, lane, wave);    // Znew = W Z (into old Y buffer)
      __syncthreads();
      float* nY = T; float* nZ = Y; float* nW = Z; float* nT = W;
      Y = nY; Z = nZ; W = nW; T = nT;
    }
    if (s + 1 < SQRT_STEPS) {
      // next sqrt: M := Y (already there), Z := I
      for (int idx = tid; idx < MAT_ELEMS; idx += 256) {
        const int r = idx >> 6, c = idx & 63;
        Z[r * LDS_LD + c] = (r == c) ? 1.0f : 0.0f;
      }
      __syncthreads();
    }
  }

  // ---- E = Y - I (spec(E) small after the repeated roots) ----
  if (tid < MATN) Y[tid * LDS_LD + tid] -= 1.0f;
  __syncthreads();

  // ---- log(I+E) via 8-term Mercator series, Horner in E ----
  // P(E) = sum_{j=1..8} c_j E^{j-1},  c_j = (-1)^(j+1)/j ;  log = E * P(E)
  float* E = Y;
  float* S = Z;   // free
  float* H = W;   // free
  for (int idx = tid; idx < MAT_ELEMS; idx += 256) {
    const int r = idx >> 6, c = idx & 63;
    float v = (-1.0f / 8.0f) * E[r * LDS_LD + c];     // c8 * E
    if (r == c) v += (1.0f / 7.0f);                   // + c7 I
    S[r * LDS_LD + c] = v;
  }
  __syncthreads();
  for (int j = 6; j >= 1; --j) {
    const float cj = ((j & 1) ? 1.0f : -1.0f) / (float)j;
    mm64(E, S, H, 1.0f, cj, lane, wave);              // S' = E S + c_j I
    __syncthreads();
    float* tmp = S; S = H; H = tmp;
  }
  // out = 2^SQRT_STEPS * (E * P(E)) + log(c) * I
  const float log_c = scale[1];
  mm64(E, S, H, LOG_SCALE, log_c, lane, wave);
  __syncthreads();

  // ---- Store ----
  for (int idx = tid; idx < MAT_ELEMS; idx += 256) {
    const int r = idx >> 6, c = idx & 63;
    out[gbase + idx] = H[r * LDS_LD + c];
  }
}

extern "C" void kernel_launch(void* const* d_in, const int* in_sizes, int n_in,
                              void* d_out, int out_size, void* d_ws, size_t ws_size,
                              hipStream_t stream) {
  (void)n_in; (void)d_ws; (void)ws_size; (void)out_size;
  const float* X = (const float*)d_in[0];
  float* out = (float*)d_out;
  const int nmat = in_sizes[0] / MAT_ELEMS;   // 8192
  logm_spd64_kernel<<<dim3(nmat), dim3(256), 0, stream>>>(X, out, nmat);
}